// _PointnetSAModuleBase_85358180040850
// MI455X (gfx1250) — compile-verified
//
#include <hip/hip_runtime.h>

#define Bx 8
#define Nn 8192
#define Sx 2048
#define NSAMP 32
#define Cx 64
#define R2 0.04f
#define BIGF 1e10f
#define LDW 264      // LDS row stride (256 cols + 8 pad)
#define TQ 8         // query points per MLP block -> 256 positions

typedef __attribute__((ext_vector_type(2))) float v2f;
typedef __attribute__((ext_vector_type(8))) float v8f;

// ---------------------------------------------------------------------------
// Kernel 1: furthest point sampling. 1 block per batch, 1024 threads,
// 8 points per thread held in registers. Sequential 2048-step argmax loop.
// ---------------------------------------------------------------------------
__global__ __launch_bounds__(1024) void fps_kernel(const float* __restrict__ xyz,
                                                   float* __restrict__ new_xyz)
{
    __shared__ float redv[32];
    __shared__ int   redi[32];
    __shared__ float sf[3];
    const int b    = blockIdx.x;
    const int tid  = threadIdx.x;
    const int w    = tid >> 5;
    const int lane = tid & 31;
    const float* base = xyz + (size_t)b * Nn * 3;

    float px[8], py[8], pz[8], dist[8];
    const int n0 = tid * 8;                    // blocked ownership keeps index order
#pragma unroll
    for (int i = 0; i < 8; ++i) {
        px[i] = base[(n0 + i) * 3 + 0];
        py[i] = base[(n0 + i) * 3 + 1];
        pz[i] = base[(n0 + i) * 3 + 2];
        dist[i] = BIGF;
    }
    int far = 0;                               // live only in thread 0
    for (int s = 0; s < Sx; ++s) {
        if (tid == 0) {
            float fx = base[far * 3 + 0];
            float fy = base[far * 3 + 1];
            float fz = base[far * 3 + 2];
            sf[0] = fx; sf[1] = fy; sf[2] = fz;
            new_xyz[((size_t)b * Sx + s) * 3 + 0] = fx;
            new_xyz[((size_t)b * Sx + s) * 3 + 1] = fy;
            new_xyz[((size_t)b * Sx + s) * 3 + 2] = fz;
        }
        __syncthreads();
        const float fx = sf[0], fy = sf[1], fz = sf[2];
        float bd = -1.0f; int bi = 0;
#pragma unroll
        for (int i = 0; i < 8; ++i) {
            float dx = px[i] - fx, dy = py[i] - fy, dz = pz[i] - fz;
            float d  = dx * dx + dy * dy + dz * dz;
            dist[i]  = fminf(dist[i], d);
            if (dist[i] > bd) { bd = dist[i]; bi = n0 + i; }   // strict > = first max
        }
#pragma unroll
        for (int off = 16; off >= 1; off >>= 1) {
            float od = __shfl_xor(bd, off, 32);
            int   oi = __shfl_xor(bi, off, 32);
            if (od > bd || (od == bd && oi < bi)) { bd = od; bi = oi; }
        }
        if (lane == 0) { redv[w] = bd; redi[w] = bi; }
        __syncthreads();
        if (w == 0) {
            float vd = redv[lane]; int vi = redi[lane];
#pragma unroll
            for (int off = 16; off >= 1; off >>= 1) {
                float od = __shfl_xor(vd, off, 32);
                int   oi = __shfl_xor(vi, off, 32);
                if (od > vd || (od == vd && oi < vi)) { vd = od; vi = oi; }
            }
            if (lane == 0) far = vi;
        }
        // redv rewrite next iter happens only after everyone passes the
        // __syncthreads at the top, which wave0 reaches after reading redv.
    }
}

// ---------------------------------------------------------------------------
// Kernel 2: ball query. One wave (32 lanes) per query point; scan the 8192
// candidates in index order, ballot + prefix-popcount to append the first 32
// in-radius hits, pad with the first hit (pointnet2 semantics).
// ---------------------------------------------------------------------------
__global__ __launch_bounds__(256) void ballq_kernel(const float* __restrict__ xyz,
                                                    const float* __restrict__ new_xyz,
                                                    int* __restrict__ idx_out)
{
    __shared__ int lists[8][NSAMP];
    const int w    = threadIdx.x >> 5;
    const int lane = threadIdx.x & 31;
    const int gid  = blockIdx.x * 8 + w;       // b*S + s
    const int b    = gid >> 11;                // /2048
    const float* base = xyz + (size_t)b * Nn * 3;
    const float qx = new_xyz[(size_t)gid * 3 + 0];
    const float qy = new_xyz[(size_t)gid * 3 + 1];
    const float qz = new_xyz[(size_t)gid * 3 + 2];

    int cnt = 0;                               // uniform across the wave
    for (int nb = 0; nb < Nn && cnt < NSAMP; nb += 32) {
        const int n = nb + lane;
        float dx = base[n * 3 + 0] - qx;
        float dy = base[n * 3 + 1] - qy;
        float dz = base[n * 3 + 2] - qz;
        float d2 = dx * dx + dy * dy + dz * dz;
        bool hit = d2 < R2;
        unsigned m = (unsigned)__ballot(hit);
        int before = __popc(m & ((1u << lane) - 1u));
        if (hit) {
            int slot = cnt + before;
            if (slot < NSAMP) lists[w][slot] = n;
        }
        cnt += __popc(m);
    }
    int v;
    if (cnt == 0)            v = 0;
    else if (lane < cnt)     v = lists[w][lane];
    else                     v = lists[w][0];
    idx_out[(size_t)gid * NSAMP + lane] = v;
}

// ---------------------------------------------------------------------------
// Kernel 3: fused gather + 3x (GEMM + bias + ReLU) + 32-wide max-pool.
// Block = 256 threads (8 waves) handles 8 query points = 256 positions.
// All intermediates live in LDS; GEMMs use v_wmma_f32_16x16x4_f32.
// LDS: bufA = 68 x 264 (input, reused for h2), bufB = 64 x 264 (h1).
// ---------------------------------------------------------------------------
__global__ __launch_bounds__(256) void mlp_kernel(const float* __restrict__ xyz,
                                                  const float* __restrict__ features,
                                                  const int*   __restrict__ idx,
                                                  const float* __restrict__ new_xyz,
                                                  const float* __restrict__ w1, const float* __restrict__ b1,
                                                  const float* __restrict__ w2, const float* __restrict__ b2,
                                                  const float* __restrict__ w3, const float* __restrict__ b3,
                                                  float* __restrict__ out_feat)
{
    extern __shared__ float smem[];
    float* bufA = smem;                // 68 rows x LDW
    float* bufB = smem + 68 * LDW;     // 64 rows x LDW

    const int b     = blockIdx.x >> 8; // 256 tiles per batch
    const int tile  = blockIdx.x & 255;
    const int s0    = tile * TQ;
    const int t     = threadIdx.x;     // column 0..255
    const int wv    = t >> 5;
    const int lane  = t & 31;
    const int lhalf = lane >> 4;       // 0: K/M low, 1: high per WMMA layout
    const int l16   = lane & 15;

    // ---- gather: column t = (query s0+t/32, sample t%32) -------------------
    {
        const int q = s0 + (t >> 5);
        const int j = t & 31;
        const int n = idx[((size_t)b * Sx + q) * NSAMP + j];
        const float* p  = xyz     + ((size_t)b * Nn + n) * 3;
        const float* qq = new_xyz + ((size_t)b * Sx + q) * 3;
        bufA[0 * LDW + t] = p[0] - qq[0];
        bufA[1 * LDW + t] = p[1] - qq[1];
        bufA[2 * LDW + t] = p[2] - qq[2];
        const float* fb = features + (size_t)b * Cx * Nn + n;
#pragma unroll 4
        for (int c = 0; c < Cx; ++c)
            bufA[(3 + c) * LDW + t] = fb[(size_t)c * Nn];
        bufA[67 * LDW + t] = 0.0f;     // zero pad row for K=68
    }
    __syncthreads();

    const v8f vzero = {0, 0, 0, 0, 0, 0, 0, 0};

    // ---- GEMM1: h1[64 x 256] = relu(W1[64 x 67] * bufA + b1) --------------
    {
        const int mt = wv >> 1, nh = wv & 1, nbase = nh * 128;
        const int mrow = mt * 16 + l16;
        float bias[8];
#pragma unroll
        for (int r = 0; r < 8; ++r) bias[r] = b1[mt * 16 + r + 8 * lhalf];
        v8f acc[8];
#pragma unroll
        for (int i = 0; i < 8; ++i) acc[i] = vzero;
        for (int kk = 0; kk < 17; ++kk) {
            const int kc = kk * 4 + lhalf * 2;
            v2f a;
            a.x = (kc     < 67) ? w1[mrow * 67 + kc]     : 0.0f;
            a.y = (kc + 1 < 67) ? w1[mrow * 67 + kc + 1] : 0.0f;
#pragma unroll
            for (int nt = 0; nt < 8; ++nt) {
                const int ncol = nbase + nt * 16 + l16;
                v2f bb;
                bb.x = bufA[kc * LDW + ncol];
                bb.y = bufA[(kc + 1) * LDW + ncol];
                acc[nt] = __builtin_amdgcn_wmma_f32_16x16x4_f32(
                    false, a, false, bb, (short)0, acc[nt], false, false);
            }
        }
#pragma unroll
        for (int nt = 0; nt < 8; ++nt)
#pragma unroll
            for (int r = 0; r < 8; ++r) {
                float v = fmaxf(acc[nt][r] + bias[r], 0.0f);
                const int m = mt * 16 + r + 8 * lhalf;
                bufB[m * LDW + nbase + nt * 16 + l16] = v;
            }
    }
    __syncthreads();

    // ---- GEMM2: h2[64 x 256] = relu(W2[64 x 64] * h1 + b2) -> bufA --------
    {
        const int mt = wv >> 1, nh = wv & 1, nbase = nh * 128;
        const int mrow = mt * 16 + l16;
        float bias[8];
#pragma unroll
        for (int r = 0; r < 8; ++r) bias[r] = b2[mt * 16 + r + 8 * lhalf];
        v8f acc[8];
#pragma unroll
        for (int i = 0; i < 8; ++i) acc[i] = vzero;
        for (int kk = 0; kk < 16; ++kk) {
            const int kc = kk * 4 + lhalf * 2;
            v2f a;
            a.x = w2[mrow * 64 + kc];
            a.y = w2[mrow * 64 + kc + 1];
#pragma unroll
            for (int nt = 0; nt < 8; ++nt) {
                const int ncol = nbase + nt * 16 + l16;
                v2f bb;
                bb.x = bufB[kc * LDW + ncol];
                bb.y = bufB[(kc + 1) * LDW + ncol];
                acc[nt] = __builtin_amdgcn_wmma_f32_16x16x4_f32(
                    false, a, false, bb, (short)0, acc[nt], false, false);
            }
        }
#pragma unroll
        for (int nt = 0; nt < 8; ++nt)
#pragma unroll
            for (int r = 0; r < 8; ++r) {
                float v = fmaxf(acc[nt][r] + bias[r], 0.0f);
                const int m = mt * 16 + r + 8 * lhalf;
                bufA[m * LDW + nbase + nt * 16 + l16] = v;
            }
    }
    __syncthreads();

    // ---- GEMM3 + max-pool: out[128 x 8] = max_ns relu(W3 * h2 + b3) -------
    {
        const int mt = wv;                   // one M-tile of 16 rows per wave
        const int mrow = mt * 16 + l16;
        float bias[8];
#pragma unroll
        for (int r = 0; r < 8; ++r) bias[r] = b3[mt * 16 + r + 8 * lhalf];
        for (int nh = 0; nh < 2; ++nh) {
            const int nbase = nh * 128;
            v8f acc[8];
#pragma unroll
            for (int i = 0; i < 8; ++i) acc[i] = vzero;
            for (int kk = 0; kk < 16; ++kk) {
                const int kc = kk * 4 + lhalf * 2;
                v2f a;
                a.x = w3[mrow * 64 + kc];
                a.y = w3[mrow * 64 + kc + 1];
#pragma unroll
                for (int nt = 0; nt < 8; ++nt) {
                    const int ncol = nbase + nt * 16 + l16;
                    v2f bb;
                    bb.x = bufA[kc * LDW + ncol];
                    bb.y = bufA[(kc + 1) * LDW + ncol];
                    acc[nt] = __builtin_amdgcn_wmma_f32_16x16x4_f32(
                        false, a, false, bb, (short)0, acc[nt], false, false);
                }
            }
            // max over each group of 32 columns (= 2 adjacent 16-col tiles)
            for (int g = 0; g < 4; ++g) {
                float mx[8];
#pragma unroll
                for (int r = 0; r < 8; ++r) {
                    float va = fmaxf(acc[2 * g][r]     + bias[r], 0.0f);
                    float vb = fmaxf(acc[2 * g + 1][r] + bias[r], 0.0f);
                    mx[r] = fmaxf(va, vb);
                }
#pragma unroll
                for (int r = 0; r < 8; ++r)
#pragma unroll
                    for (int off = 1; off < 16; off <<= 1)
                        mx[r] = fmaxf(mx[r], __shfl_xor(mx[r], off, 32));
                const int sidx = s0 + nh * 4 + g;
                if (l16 == 0) {              // lanes 0 and 16 hold the results
#pragma unroll
                    for (int r = 0; r < 8; ++r) {
                        const int m = mt * 16 + r + 8 * lhalf;
                        out_feat[(size_t)b * 128 * Sx + (size_t)m * Sx + sidx] = mx[r];
                    }
                }
            }
        }
    }
}

// ---------------------------------------------------------------------------
extern "C" void kernel_launch(void* const* d_in, const int* in_sizes, int n_in,
                              void* d_out, int out_size, void* d_ws, size_t ws_size,
                              hipStream_t stream)
{
    const float* xyz  = (const float*)d_in[0];
    const float* feat = (const float*)d_in[1];
    const float* w1   = (const float*)d_in[2];
    const float* b1   = (const float*)d_in[3];
    const float* w2   = (const float*)d_in[4];
    const float* b2   = (const float*)d_in[5];
    const float* w3   = (const float*)d_in[6];
    const float* b3   = (const float*)d_in[7];

    float* out      = (float*)d_out;
    float* new_xyz  = out;                          // (8,2048,3)  output 0
    float* out_feat = out + (size_t)Bx * Sx * 3;    // (8,128,2048) output 1
    int*   idx      = (int*)d_ws;                   // (8,2048,32) scratch

    fps_kernel<<<Bx, 1024, 0, stream>>>(xyz, new_xyz);
    ballq_kernel<<<(Bx * Sx) / 8, 256, 0, stream>>>(xyz, new_xyz, idx);

    const size_t smem = (size_t)(68 + 64) * LDW * sizeof(float);  // ~136 KB
    (void)hipFuncSetAttribute(reinterpret_cast<const void*>(mlp_kernel),
                              hipFuncAttributeMaxDynamicSharedMemorySize, (int)smem);
    mlp_kernel<<<Bx * (Sx / TQ), 256, smem, stream>>>(
        xyz, feat, idx, new_xyz, w1, b1, w2, b2, w3, b3, out_feat);
}